// Net_82703890252600
// MI455X (gfx1250) — compile-verified
//
#include <hip/hip_runtime.h>

// GNN message passing for MI455X (gfx1250), fp32 end-to-end via
// V_WMMA_F32_16X16X4_F32 (wave32 WMMA). A-tiles double-buffered in LDS via
// CDNA5 async global->LDS copies (ASYNCcnt); weights register-resident per
// wave; edge aggregation via native f32 L2 atomics. Everything is a uniform
// K=128 GEMM with optional accumulate-in, so K=256 contractions chain two
// launches through an L2-resident accumulator.

#define N_NODES 50000
#define N_EDGES 800000
#define DIM     128
#define NLAYERS 21
#define OUTD    63

typedef __attribute__((ext_vector_type(2))) float v2f;
typedef __attribute__((ext_vector_type(8))) float v8f;

#if __has_builtin(__builtin_amdgcn_global_load_async_to_lds_b128) && \
    __has_builtin(__builtin_amdgcn_s_wait_asynccnt)
#define USE_ASYNC_LDS 1
typedef int v4i __attribute__((vector_size(16)));
typedef __attribute__((address_space(1))) v4i as1_v4i;  // global
typedef __attribute__((address_space(3))) v4i as3_v4i;  // LDS
#endif

// Copy 16 bytes global -> LDS. Async path: hardware copy tracked by ASYNCcnt
// (no VGPR round-trip). Generic->AS3: low 32 bits of a generic LDS pointer
// are the LDS byte offset (ISA 10.2), so the integer truncation is exact.
__device__ __forceinline__ void cp16_to_lds(const float* gp, float* lp) {
#ifdef USE_ASYNC_LDS
  __builtin_amdgcn_global_load_async_to_lds_b128(
      (as1_v4i*)(unsigned long long)gp, (as3_v4i*)(unsigned long long)lp, 0, 0);
#else
  *(float4*)lp = *(const float4*)gp;
#endif
}

__device__ __forceinline__ void wait_stage() {
#ifdef USE_ASYNC_LDS
  __builtin_amdgcn_s_wait_asynccnt(0);
#endif
}

__device__ __forceinline__ v8f wmma_f32_16x16x4(v2f a, v2f b, v8f c) {
  // (neg_a, A, neg_b, B, c_mod, C, reuse_a, reuse_b)
  return __builtin_amdgcn_wmma_f32_16x16x4_f32(false, a, false, b, 0, c, false, false);
}

// ---------------------------------------------------------------------------
// K=128 GEMM: C[n, coloff+j] = act( A[n,:]·W[j,:] + bias[j] (+ Cin[n,coloff+j]) )
//   A: N_NODES rows, row stride ald, 128 cols used.  W: rows stride wld.
//   Each wave owns one 16-col tile (NTILES <= 8); its 16 W rows live in VGPRs
//   for the whole kernel. 16-row A-tiles are staged into double-buffered LDS
//   by all 256 threads (async), then read as 64-bit fragments (row pad of 4
//   floats -> bank = 4*row+col -> conflict-free) and consumed by a burst of
//   32 chained v_wmma_f32_16x16x4_f32.
// ACT: 0 none, 1 relu, 2 tanh.  COPY2: mirror the result into C2.
// ---------------------------------------------------------------------------
template <int ACT, bool COPY2, bool ACCIN, int NTILES>
__global__ __launch_bounds__(256) void gemm128_kernel(
    const float* __restrict__ A, int ald,
    const float* __restrict__ W, int wld,
    const float* __restrict__ bias,
    const float* __restrict__ Cin,
    float* __restrict__ C, float* __restrict__ C2,
    int ldc, int coloff, int nvalid) {
  constexpr int KP = DIM + 4;  // padded LDS row stride (floats)
  __shared__ float smem[2][16 * KP];
  const int wave  = threadIdx.x >> 5;
  const int lane  = threadIdx.x & 31;
  const int frow  = lane & 15;  // M index (A) / N index (B, C/D) within tile
  const int khalf = lane >> 4;  // K pair {k0,k0+1} vs {k0+2,k0+3}
  const int NRB   = N_NODES / 16;

  const bool is_compute = (wave < NTILES);
  const int n0   = wave * 16;
  const int wrow = n0 + frow;
  const bool wvalid = is_compute && (wrow < nvalid);

  // Per-wave register-resident B fragments (16 W rows x 128 K).
  v2f bfrag[32];
  float bval = 0.f;
  if (is_compute) {
#pragma unroll
    for (int ks = 0; ks < 32; ++ks) {
      v2f z = {0.f, 0.f};
      bfrag[ks] =
          wvalid ? *(const v2f*)(W + (size_t)wrow * wld + 4 * ks + 2 * khalf)
                 : z;
    }
    if (bias != nullptr && wvalid) bval = bias[wrow];
  }

  // Cooperative tile stage: 16 rows x 512B = 512 chunks of 16B, 2 per thread.
  auto stage = [&](int rb, float* buf) {
    const int row0 = rb * 16;
#pragma unroll
    for (int it = 0; it < 2; ++it) {
      const int c    = threadIdx.x + it * 256;
      const int row  = c >> 5;
      const int colf = (c & 31) * 4;
      cp16_to_lds(A + (size_t)(row0 + row) * ald + colf, buf + row * KP + colf);
    }
  };

  int rb = blockIdx.x;
  if (rb < NRB) stage(rb, smem[0]);
  wait_stage();
  __syncthreads();

  int cur = 0;
  for (; rb < NRB; rb += gridDim.x) {
    const int rbn = rb + gridDim.x;
    if (rbn < NRB) stage(rbn, smem[cur ^ 1]);  // overlap with compute below

    if (is_compute) {
      const int row0 = rb * 16;
      const float* buf = &smem[cur][0];
      v8f acc;
#pragma unroll
      for (int r = 0; r < 8; ++r) {
        float v = bval;
        if (ACCIN)
          v += Cin[(size_t)(row0 + r + 8 * khalf) * ldc + coloff + n0 + frow];
        acc[r] = v;
      }
#pragma unroll
      for (int half = 0; half < 2; ++half) {
        v2f af[16];
#pragma unroll
        for (int j = 0; j < 16; ++j)
          af[j] = *(const v2f*)(buf + frow * KP + 64 * half + 4 * j + 2 * khalf);
#pragma unroll
        for (int j = 0; j < 16; ++j)
          acc = wmma_f32_16x16x4(af[j], bfrag[16 * half + j], acc);
      }
#pragma unroll
      for (int r = 0; r < 8; ++r) {
        float v = acc[r];
        if (ACT == 1) v = fmaxf(v, 0.f);
        if (ACT == 2) v = tanhf(v);
        const int row = row0 + r + 8 * khalf;
        const int col = n0 + frow;
        if (col < nvalid) {
          C[(size_t)row * ldc + coloff + col] = v;
          if (COPY2) C2[(size_t)row * ldc + coloff + col] = v;
        }
      }
    }
    wait_stage();      // own async copies issued this iteration
    __syncthreads();   // all waves' copies + reads of smem[cur] complete
    cur ^= 1;
  }
}

// ---------------------------------------------------------------------------
// Edge scatter: aggr[dst[e], :] += m[src[e], :]  (one wave per edge, float4
// per lane, native f32 atomics resolved in L2)
// ---------------------------------------------------------------------------
__global__ __launch_bounds__(256) void scatter_kernel(
    const int* __restrict__ ei, const float* __restrict__ m,
    float* __restrict__ aggr) {
  const int e = blockIdx.x * 8 + (threadIdx.x >> 5);
  if (e >= N_EDGES) return;
  const int lane = threadIdx.x & 31;
  const int s = ei[e];
  const int d = ei[N_EDGES + e];
  const float4 v = *(const float4*)(m + (size_t)s * DIM + lane * 4);
  float* p = aggr + (size_t)d * DIM + lane * 4;
  unsafeAtomicAdd(p + 0, v.x);
  unsafeAtomicAdd(p + 1, v.y);
  unsafeAtomicAdd(p + 2, v.z);
  unsafeAtomicAdd(p + 3, v.w);
}

__global__ __launch_bounds__(256) void copy_kernel(
    const float* __restrict__ src, float* __restrict__ dst, int n4) {
  int i = blockIdx.x * blockDim.x + threadIdx.x;
  const int stride = gridDim.x * blockDim.x;
  for (; i < n4; i += stride) {
    ((float4*)dst)[i] = ((const float4*)src)[i];
  }
}

extern "C" void kernel_launch(void* const* d_in, const int* in_sizes, int n_in,
                              void* d_out, int out_size, void* d_ws,
                              size_t ws_size, hipStream_t stream) {
  (void)in_sizes; (void)n_in; (void)out_size; (void)ws_size;
  const float* x     = (const float*)d_in[0];
  const int*   ei    = (const int*)d_in[1];    // [2, E] (src row, dst row)
  const float* W_msg = (const float*)d_in[3];  // [21,128,128]
  const float* b_msg = (const float*)d_in[4];  // [21,128]
  const float* W_upd = (const float*)d_in[5];  // [21,128,256]
  const float* W1    = (const float*)d_in[6];  // [256,128]
  const float* b1    = (const float*)d_in[7];
  const float* W2    = (const float*)d_in[8];  // [128,256]
  const float* b2    = (const float*)d_in[9];
  const float* W3    = (const float*)d_in[10]; // [63,128]
  const float* b3    = (const float*)d_in[11];
  float* out = (float*)d_out;

  float* h    = (float*)d_ws;                    // N x 128
  float* m    = h    + (size_t)N_NODES * DIM;    // N x 128
  float* aggr = m    + (size_t)N_NODES * DIM;    // N x 128
  float* t256 = aggr + (size_t)N_NODES * DIM;    // N x 256
  float* tacc = t256 + (size_t)N_NODES * 2 * DIM;  // N x 128 accumulator

  copy_kernel<<<1024, 256, 0, stream>>>(x, h, N_NODES * DIM / 4);

  const int GG = 1024;  // grid-stride over 3125 node row-blocks
  for (int l = 0; l < NLAYERS; ++l) {
    const float* Wm = W_msg + (size_t)l * DIM * DIM;
    const float* bm = b_msg + (size_t)l * DIM;
    const float* Wu = W_upd + (size_t)l * DIM * 2 * DIM;  // [128 x 256]
    // m = tanh(h @ Wm^T + bm); aggr initialized to m (self term)
    gemm128_kernel<2, true, false, 8><<<GG, 256, 0, stream>>>(
        h, DIM, Wm, DIM, bm, nullptr, m, aggr, DIM, 0, DIM);
    // aggr[dst] += m[src]
    scatter_kernel<<<N_EDGES / 8, 256, 0, stream>>>(ei, m, aggr);
    // concat-GEMM as two K=128 passes: tacc = aggr @ Wu[:, :128]^T
    gemm128_kernel<0, false, false, 8><<<GG, 256, 0, stream>>>(
        aggr, DIM, Wu, 2 * DIM, nullptr, nullptr, tacc, nullptr, DIM, 0, DIM);
    // h' = relu(h @ Wu[:, 128:]^T + tacc)  -> m, then swap
    gemm128_kernel<1, false, true, 8><<<GG, 256, 0, stream>>>(
        h, DIM, Wu + DIM, 2 * DIM, nullptr, tacc, m, nullptr, DIM, 0, DIM);
    float* t = h; h = m; m = t;
  }

  // MLP head. W1: 256x128 -> two column-half launches.
  gemm128_kernel<1, false, false, 8><<<GG, 256, 0, stream>>>(
      h, DIM, W1, DIM, b1, nullptr, t256, nullptr, 2 * DIM, 0, DIM);
  gemm128_kernel<1, false, false, 8><<<GG, 256, 0, stream>>>(
      h, DIM, W1 + (size_t)DIM * DIM, DIM, b1 + DIM, nullptr,
      t256, nullptr, 2 * DIM, DIM, DIM);
  // W2: 128x256 -> two K=128 passes chained through tacc.
  gemm128_kernel<0, false, false, 8><<<GG, 256, 0, stream>>>(
      t256, 2 * DIM, W2, 2 * DIM, nullptr, nullptr, tacc, nullptr, DIM, 0, DIM);
  gemm128_kernel<1, false, true, 8><<<GG, 256, 0, stream>>>(
      t256 + DIM, 2 * DIM, W2 + DIM, 2 * DIM, b2, tacc, m, nullptr, DIM, 0, DIM);
  // W3: 63x128 output head (4 column tiles, columns < 63 stored).
  gemm128_kernel<0, false, false, 4><<<GG, 256, 0, stream>>>(
      m, DIM, W3, DIM, b3, nullptr, out, nullptr, OUTD, 0, OUTD);
}